// LengthRegulator_26577257627700
// MI455X (gfx1250) — compile-verified
//
#include <hip/hip_runtime.h>

typedef __attribute__((ext_vector_type(2))) float v2f;
typedef __attribute__((ext_vector_type(4))) float v4f;
typedef __attribute__((ext_vector_type(8))) float v8f;

constexpr int kB = 32, kC = 256, kTin = 256, kTout = 2048;
constexpr int TBLK   = 128;        // T_out columns per workgroup
constexpr int NT     = TBLK / 16;  // 8 N-tiles per wave
constexpr int KSTAGE = 64;         // K rows of path staged per pass
constexpr int NKC    = KSTAGE / 4; // 16 K-chunks per stage
constexpr int GROUP  = 80;         // floats per (kc,nt) fragment group (64 data + 16 pad)
constexpr int STAGEF = NKC * NT * GROUP; // 10240 floats = 40 KB per stage

__global__ __launch_bounds__(256)
void lr_bmm_wmma(const float* __restrict__ x,
                 const float* __restrict__ path,
                 float* __restrict__ out) {
  __shared__ float sp[2 * STAGEF]; // 81920 bytes, double-buffered

  const int tid  = threadIdx.x;
  const int wave = tid >> 5;   // 0..7
  const int lane = tid & 31;
  const int tl   = lane & 15;  // N / M index within tile
  const int h    = lane >> 4;  // half-wave select

  const int b  = blockIdx.x >> 4;          // batch
  const int tb = (blockIdx.x & 15) * TBLK; // T_out panel base

  // each wave owns C rows [wave*32, wave*32+32): two 16-row M-tiles
  const int crow0 = wave * 32 + tl;
  const float* xrow0 = x + ((size_t)b * kC + crow0) * kTin;
  const float* xrow1 = xrow0 + (size_t)16 * kTin;
  const float* pbase = path + (size_t)b * kTin * kTout + tb;

  // ---- staging decomposition, fully hoisted to per-thread constants ----
  // chunk id c = it*256 + tid  ->  kc = it*2 + tid[7], h = tid[6], u = tid[5:0]
  const int skc = tid >> 7;        // kc base (0/1)
  const int shh = (tid >> 6) & 1;  // half-wave row-pair select
  const int st2 = (tid & 63) * 2;  // even local col 0..126
  const int snt = st2 >> 4;
  const int stt = st2 & 15;
  const float* sgbase = pbase + (size_t)(skc * 4 + 2 * shh) * kTout + st2;
  const int    slbase = (skc * NT + snt) * GROUP + (shh * 16 + stt) * 2;

  // lds[(kc*NT+nt)*GROUP + (h*16+tl)*2 + p] = path[kc*4 + 2h + p][nt*16 + tl]
  auto stage = [&](int ks, float* sbuf) {
    const float* g = sgbase + (size_t)ks * KSTAGE * kTout; // all it-offsets are immediates
    float* l = sbuf + slbase;
#pragma unroll
    for (int it = 0; it < 8; ++it) {
      v2f g0 = *(const v2f*)(g + (size_t)it * 8 * kTout);          // row k,   coalesced
      v2f g1 = *(const v2f*)(g + (size_t)it * 8 * kTout + kTout);  // row k+1
      v4f w;
      w.x = g0.x; w.y = g1.x; w.z = g0.y; w.w = g1.y; // interleave to fragment order
      *(v4f*)(l + it * 2 * NT * GROUP) = w;           // ds_store_b128, imm offset
    }
  };

  v8f acc[2][NT];

  auto body = [&](const float* cur, int ks, int kc) {
    const int kk = ks * KSTAGE + kc * 4 + 2 * h;
    // A fragments: 16x4 f32 layout -> lanes 0-15: K=0,1 ; lanes 16-31: K=2,3
    v2f a0 = *(const v2f*)(xrow0 + kk);
    v2f a1 = *(const v2f*)(xrow1 + kk);
    const float* lg = cur + kc * (NT * GROUP) + (h * 16 + tl) * 2;
#pragma unroll
    for (int nt = 0; nt < NT; ++nt) {
      v2f bf = *(const v2f*)(lg + nt * GROUP); // one ds_load_b64, conflict-free
      acc[0][nt] = __builtin_amdgcn_wmma_f32_16x16x4_f32(
          false, a0, false, bf, (short)0, acc[0][nt], false, false);
      acc[1][nt] = __builtin_amdgcn_wmma_f32_16x16x4_f32(
          false, a1, false, bf, (short)0, acc[1][nt], false, false);
    }
  };

  stage(0, sp);
  __syncthreads();
  stage(1, sp + STAGEF); // overlap next stage with ks=0 compute

  // ---- ks = 0, kc = 0 peeled: accumulators born from WMMA with C = 0 ----
  {
    const v8f z = (v8f){0.f, 0.f, 0.f, 0.f, 0.f, 0.f, 0.f, 0.f};
    v2f a0 = *(const v2f*)(xrow0 + 2 * h);
    v2f a1 = *(const v2f*)(xrow1 + 2 * h);
    const float* lg = sp + (h * 16 + tl) * 2;
#pragma unroll
    for (int nt = 0; nt < NT; ++nt) {
      v2f bf = *(const v2f*)(lg + nt * GROUP);
      acc[0][nt] = __builtin_amdgcn_wmma_f32_16x16x4_f32(
          false, a0, false, bf, (short)0, z, false, false);
      acc[1][nt] = __builtin_amdgcn_wmma_f32_16x16x4_f32(
          false, a1, false, bf, (short)0, z, false, false);
    }
  }
#pragma unroll 5
  for (int kc = 1; kc < NKC; ++kc) body(sp, 0, kc);
  __syncthreads();

  for (int ks = 1; ks < kTin / KSTAGE; ++ks) {
    const float* cur = sp + (ks & 1) * STAGEF;
    if (ks + 1 < kTin / KSTAGE)
      stage(ks + 1, sp + ((ks + 1) & 1) * STAGEF); // overlap with compute below
#pragma unroll 4
    for (int kc = 0; kc < NKC; ++kc) body(cur, ks, kc);
    __syncthreads();
  }

  // D layout: VGPR r -> lanes 0-15: M=r, lanes 16-31: M=r+8; N = lane&15
#pragma unroll
  for (int m = 0; m < 2; ++m) {
    float* obase = out + ((size_t)b * kC + wave * 32 + m * 16 + h * 8) * kTout + tb + tl;
#pragma unroll
    for (int nt = 0; nt < NT; ++nt) {
#pragma unroll
      for (int rr = 0; rr < 8; ++rr) {
        __builtin_nontemporal_store(acc[m][nt][rr],
                                    obase + (size_t)rr * kTout + nt * 16);
      }
    }
  }
}

extern "C" void kernel_launch(void* const* d_in, const int* in_sizes, int n_in,
                              void* d_out, int out_size, void* d_ws, size_t ws_size,
                              hipStream_t stream) {
  (void)in_sizes; (void)n_in; (void)out_size; (void)d_ws; (void)ws_size;
  const float* x    = (const float*)d_in[0];
  const float* path = (const float*)d_in[1];
  float*       out  = (float*)d_out;

  const int grid = kB * (kTout / TBLK); // 32 * 16 = 512 workgroups
  lr_bmm_wmma<<<grid, 256, 0, stream>>>(x, path, out);
}